// SpatialEdge_44667659879032
// MI455X (gfx1250) — compile-verified
//
#include <hip/hip_runtime.h>

#define KNN      32
#define CAP      64
#define WAVES    4
#define RADIUS2  25.0f
#define MIN_DIST 5
#define GNUM     12

typedef __attribute__((ext_vector_type(2))) float v2f;
typedef __attribute__((ext_vector_type(8))) float v8f;

// ---------------------------------------------------------------------------
// Kernel 1: graph segment boundaries (node2graph is sorted) + scalar output.
// starts[g] = lower_bound(node2graph, g); starts[GNUM] = N.
// ---------------------------------------------------------------------------
__global__ void se_segments(const int* __restrict__ n2g, int n,
                            int* __restrict__ starts,
                            float* __restrict__ out, long long scalar_off) {
    int g = threadIdx.x;
    if (g <= GNUM) {
        int lo = 0, hi = n;
        while (lo < hi) {
            int mid = (lo + hi) >> 1;
            if (n2g[mid] < g) lo = mid + 1; else hi = mid;
        }
        starts[g] = lo;
    }
    if (threadIdx.x == 0) out[scalar_off] = 1.0f;
}

// ---------------------------------------------------------------------------
// Kernel 2: one wave per 16-center tile.
//   d2(i,j) = |pi|^2 + |pj|^2 - 2 pi.pj  computed as a 16x16x4 f32 WMMA:
//   A row m = (-2x,-2y,-2z, 1), B col n = (x,y,z,|p|^2), C(m,n) = |p_m|^2.
// Candidates within radius are ballot-compacted into per-row LDS lists,
// then ranked exactly (ascending d2, ties by index == jax top_k order).
// ---------------------------------------------------------------------------
__global__ void __launch_bounds__(WAVES * 32)
se_knn(const float* __restrict__ pos, const int* __restrict__ n2g,
       const int* __restrict__ res, const int* __restrict__ starts,
       float* __restrict__ out, int n) {
    __shared__ float sD2 [WAVES * 16 * CAP];
    __shared__ int   sIdx[WAVES * 16 * CAP];
    __shared__ int   sCnt[WAVES * 16];
    __shared__ int   sStage[WAVES * KNN];

    const int wave = threadIdx.x >> 5;
    const int lane = threadIdx.x & 31;
    const int half = lane >> 4;        // 0: lanes 0-15, 1: lanes 16-31
    const int l16  = lane & 15;
    const int i0   = (blockIdx.x * WAVES + wave) * 16;
    if (i0 >= n) return;

    // ---- center tile (A and C operands) ----
    const int icL = min(i0 + l16, n - 1);
    const float cx = pos[icL * 3 + 0];
    const float cy = pos[icL * 3 + 1];
    const float cz = pos[icL * 3 + 2];
    const float csq = cx * cx + cy * cy + cz * cz;
    const int   cg  = n2g[icL];

    // A 16x4 layout: lanes 0-15 hold K=0,1; lanes 16-31 hold K=2,3 (same M=l16)
    v2f a;
    if (half == 0) { a.x = -2.0f * cx; a.y = -2.0f * cy; }
    else           { a.x = -2.0f * cz; a.y = 1.0f; }

    // C 16x16 f32 layout: VGPR r holds row r (lanes 0-15) / row r+8 (lanes 16-31)
    v8f cmat;
#pragma unroll
    for (int r = 0; r < 8; ++r)
        cmat[r] = __shfl(csq, r + half * 8, 32);

    // Hoisted per-lane row constants (loop-invariant over candidate tiles):
    // this lane evaluates row (r + half*8) at unroll step r.
    int gcRow[8];   // graph id of that row's center (-2 if row past N)
    int icRow[8];   // that row's center index
#pragma unroll
    for (int r = 0; r < 8; ++r) {
        const int row = r + half * 8;
        icRow[r] = i0 + row;
        const int g = __shfl(cg, row, 32);
        gcRow[r] = (icRow[r] < n) ? g : -2;
    }

    int cnt[16];
#pragma unroll
    for (int m = 0; m < 16; ++m) cnt[m] = 0;

    const int gFirst = n2g[i0];
    const int gLast  = n2g[min(i0 + 15, n - 1)];
    const int segS   = starts[gFirst];
    const int segE   = starts[gLast + 1];
    const int cbase  = (wave * 16) * CAP;

    const unsigned lsub       = half ? (unsigned)(lane - 16) : (unsigned)lane;
    const unsigned laneMaskLt = (1u << lsub) - 1u;

    // ---- scan candidate segment in 16-wide tiles ----
    for (int j0 = segS; j0 < segE; j0 += 16) {
        const int  j   = j0 + l16;
        const bool inb = (j < segE);
        const int  jc  = inb ? j : (segE - 1);
        const float bx = pos[jc * 3 + 0];
        const float by = pos[jc * 3 + 1];
        const float bz = pos[jc * 3 + 2];
        const float bsq = bx * bx + by * by + bz * bz;
        const int   gj  = inb ? n2g[jc] : -1;   // -1 never matches gcRow

        // B 4x16 layout mirrors A: lanes 0-15 hold K=0,1; lanes 16-31 hold K=2,3
        v2f b;
        if (half == 0) { b.x = bx; b.y = by; }
        else           { b.x = bz; b.y = bsq; }

        v8f d = __builtin_amdgcn_wmma_f32_16x16x4_f32(
            /*neg_a=*/false, a, /*neg_b=*/false, b,
            /*c_mod=*/(short)0, cmat, /*reuse_a=*/false, /*reuse_b=*/false);

#pragma unroll
        for (int r = 0; r < 8; ++r) {
            const float d2 = d[r];
            const bool pred = (gj == gcRow[r]) && (j != icRow[r]) &&
                              (d2 <= RADIUS2);
            const unsigned mask = (unsigned)__ballot(pred);
            const unsigned mLo  = mask & 0xFFFFu;
            const unsigned mHi  = mask >> 16;
            const unsigned sub  = half ? mHi : mLo;
            const int slot = (half ? cnt[r + 8] : cnt[r]) +
                             __popc(sub & laneMaskLt);
            if (pred && slot < CAP) {
                const int row = r + half * 8;
                sD2 [cbase + row * CAP + slot] = (d2 < 0.0f) ? 0.0f : d2;
                sIdx[cbase + row * CAP + slot] = j;
            }
            cnt[r]     += __popc(mLo);
            cnt[r + 8] += __popc(mHi);
        }
    }

    if (lane == 0) {
#pragma unroll
        for (int m = 0; m < 16; ++m)
            sCnt[wave * 16 + m] = (cnt[m] > CAP) ? CAP : cnt[m];
    }
    __asm__ volatile("s_wait_dscnt 0x0" ::: "memory");

    // ---- exact top-K by rank + filters + output ----
    const long long NK3 = (long long)n * KNN * 3;
    for (int m = 0; m < 16; ++m) {
        const int ic = i0 + m;
        if (ic >= n) break;
        const int c = sCnt[wave * 16 + m];
        sStage[wave * KNN + lane] = -1;                 // init all K slots
        __asm__ volatile("s_wait_dscnt 0x0" ::: "memory");
        const int resI = res[ic];
        for (int t = lane; t < c; t += 32) {
            const float d2t = sD2 [cbase + m * CAP + t];
            const int   jt  = sIdx[cbase + m * CAP + t];
            int rank = 0;
            for (int u = 0; u < c; ++u) {
                const float d2u = sD2 [cbase + m * CAP + u];
                const int   ju  = sIdx[cbase + m * CAP + u];
                rank += (d2u < d2t) || (d2u == d2t && ju < jt);
            }
            if (rank < KNN) {
                const int  rd = res[jt] - resI;
                const bool ok = ((rd < 0 ? -rd : rd) >= MIN_DIST) &&
                                (d2t >= 1e-20f);
                sStage[wave * KNN + rank] = ok ? jt : -1;
            }
        }
        __asm__ volatile("s_wait_dscnt 0x0" ::: "memory");
        const int jv = sStage[wave * KNN + lane];
        const long long e = ((long long)ic * KNN + lane) * 3;
        if (jv >= 0) {
            out[e + 0] = (float)jv;
            out[e + 1] = (float)ic;
            out[e + 2] = 0.0f;
            out[NK3 + (long long)ic * KNN + lane] = 1.0f;
        } else {
            out[e + 0] = -1.0f;
            out[e + 1] = -1.0f;
            out[e + 2] = -1.0f;
            out[NK3 + (long long)ic * KNN + lane] = 0.0f;
        }
    }
}

// ---------------------------------------------------------------------------
extern "C" void kernel_launch(void* const* d_in, const int* in_sizes, int n_in,
                              void* d_out, int out_size, void* d_ws, size_t ws_size,
                              hipStream_t stream) {
    const float* pos = (const float*)d_in[0];   // node_position [N,3]
    const int*   n2g = (const int*)d_in[1];     // node2graph    [N] (sorted)
    const int*   res = (const int*)d_in[2];     // atom2residue  [N]
    float* out = (float*)d_out;
    const int n = in_sizes[1];

    int* starts = (int*)d_ws;                   // GNUM+1 ints
    const long long scalar_off = (long long)n * KNN * 4;

    se_segments<<<1, 32, 0, stream>>>(n2g, n, starts, out, scalar_off);

    const int centersPerBlock = WAVES * 16;
    const int blocks = (n + centersPerBlock - 1) / centersPerBlock;
    se_knn<<<blocks, WAVES * 32, 0, stream>>>(pos, n2g, res, starts, out, n);
}